// DiffusionLayer_54520314855430
// MI455X (gfx1250) — compile-verified
//
#include <hip/hip_runtime.h>
#include <stdint.h>

// ---------------------------------------------------------------------------
// ADI diffusion (Thomas solves) on MI455X / gfx1250.
//  * 1 workgroup == 1 image (256x256 f32) resident in LDS (pitch 257 -> no
//    bank conflicts for row or column line walks, thanks to TDM pad feature).
//  * Thomas coefficients are data-independent -> precompute per-16-block
//    affine operators (lower/upper triangular 16x16 + carry vector) once,
//    then every sweep is 16-line x 16x16x16 GEMMs on the matrix pipe:
//    chained V_WMMA_F32_16X16X4_F32, carry seeded into C via outer product,
//    inter-block carry broadcast with ds_bpermute (lane 15 fwd / lane 0 bwd).
//  * HBM traffic: 33.5MB in (one tensor_load_to_lds per WG) + 33.5MB out.
// ---------------------------------------------------------------------------

#define SIZE 256
#define PITCH 257               // 256 + 1 pad dword per row
#define NSTEP 10
#define DT 0.01f
#define EPS_C 1e-06f

#define OPSZ 272                // per-block operator: 16x16 matrix + 16 carry
#define OPSET (16 * OPSZ)       // 4352 floats per coefficient set
#define NSET 21                 // 11 X-sets (k=0..10), 10 Y-sets (k=0..9)
#define OPB_OFF (NSET * OPSET)          // ws layout: [opF][opB][scal]
#define SCAL_OFF (2 * NSET * OPSET)
#define SMEM_FLOATS (SIZE * PITCH + 2 * OPSET)
#define SMEM_BYTES (SMEM_FLOATS * 4)    // 297,984 B  (< 320KB WGP LDS)

typedef __attribute__((ext_vector_type(2))) float v2f;
typedef __attribute__((ext_vector_type(8))) float v8f;
typedef __attribute__((ext_vector_type(4))) unsigned v4u;
typedef __attribute__((ext_vector_type(4))) int v4i;
typedef __attribute__((ext_vector_type(8))) int v8i;
typedef __attribute__((address_space(3))) float f32as3;

__device__ __forceinline__ float bperm(int byteIdx, float v) {
  return __builtin_bit_cast(
      float, __builtin_amdgcn_ds_bpermute(byteIdx, __builtin_bit_cast(int, v)));
}

// ---------------------------------------------------------------------------
// Setup 1: per-set scalar Thomas prep: am[i]=a_i/m_i, inv_m[i]=1/m_i, cp[i].
// Sets 0..10  : X direction, t = k*DT,        dt = DT/2
// Sets 11..20 : Y direction, t = (k+0.5)*DT,  dt = DT
// ---------------------------------------------------------------------------
__global__ void setup_scal(const float* __restrict__ abx,
                           const float* __restrict__ atcx,
                           const float* __restrict__ atqx,
                           const float* __restrict__ bby,
                           const float* __restrict__ btcy,
                           const float* __restrict__ btqy,
                           float* __restrict__ ws) {
  const int s = blockIdx.x * blockDim.x + threadIdx.x;
  if (s >= NSET) return;
  const bool isX = s < 11;
  const int k = isX ? s : s - 11;
  const float t = isX ? (float)k * DT : ((float)k + 0.5f) * DT;
  const float dtf = isX ? 0.5f * DT : DT;  // DX = DY = 1
  const float* base = isX ? abx : bby;
  const float* lin  = isX ? atcx : btcy;
  const float* quad = isX ? atqx : btqy;
  float* scal = ws + SCAL_OFF + s * 768;

  auto raw = [&](int i) {
    if (i < 0) i = 0;
    if (i > SIZE - 1) i = SIZE - 1;
    return fmaxf(base[i] + lin[i] * t + quad[i] * (t * t), EPS_C);
  };

  float cp_prev = 0.f;
  float rm1 = raw(0), r0 = raw(0);
  for (int i = 0; i < SIZE; ++i) {
    const float r1 = raw(i + 1);
    const float sm = 0.25f * rm1 + 0.5f * r0 + 0.25f * r1;  // _smooth
    const float co = sm * dtf;
    const float a = (i == 0) ? 0.f : -co;
    const float cc = (i == SIZE - 1) ? 0.f : -co;
    float bb = 1.f + 2.f * co;
    if (i == 0 || i == SIZE - 1) bb = 1.f + co;
    const float mm = bb - a * cp_prev;
    const float inv = 1.f / mm;
    const float cp = cc * inv;
    scal[i]       = a * inv;  // am
    scal[256 + i] = inv;      // inv_m
    scal[512 + i] = cp;       // c'
    cp_prev = cp;
    rm1 = r0; r0 = r1;
  }
}

// ---------------------------------------------------------------------------
// Setup 2: block operators.
//  fwd:  dp_i = sum_{j<=i} M[i][j] d_j + A[i]*carry ;
//        M[i][j] = (prod_{k=j+1..i} -am[j0+k]) * inv_m[j0+j],  A[i]=prod_{k<=i}(-am)
//  bwd:  x_i  = sum_{j>=i} N[i][j] dp_j + C[i]*carry ;
//        N[i][j] = prod_{k=i..j-1} (-cp[j0+k]),  C[i]=prod_{k=i..15}(-cp)
//  Stored transposed (op[j*16+i] = Op[i][j]) to match the WMMA B-operand
//  lane layout: VGPR p, lane half h -> K = 4*slice + p + 2h, N = lane&15.
// ---------------------------------------------------------------------------
__global__ void setup_ops(float* __restrict__ ws) {
  const int tsk = blockIdx.x * blockDim.x + threadIdx.x;
  if (tsk >= NSET * 16) return;
  const int s = tsk >> 4, b = tsk & 15;
  const float* scal = ws + SCAL_OFF + s * 768;
  const float* am  = scal;
  const float* inv = scal + 256;
  const float* cp  = scal + 512;
  const int j0 = b * 16;
  float* oF = ws + (size_t)tsk * OPSZ;
  float* oB = ws + OPB_OFF + (size_t)tsk * OPSZ;

  float A = 1.f;
  for (int i = 0; i < 16; ++i) {
    oF[i * 16 + i] = inv[j0 + i];
    float pr = 1.f;
    for (int j = i - 1; j >= 0; --j) {
      pr *= -am[j0 + j + 1];
      oF[j * 16 + i] = pr * inv[j0 + j];
    }
    for (int j = i + 1; j < 16; ++j) oF[j * 16 + i] = 0.f;
    A *= -am[j0 + i];           // block 0: am[0]==0 -> A==0 (no carry, exact)
    oF[256 + i] = A;
  }
  for (int i = 0; i < 16; ++i) {
    oB[i * 16 + i] = 1.f;
    float pr = 1.f;
    for (int j = i + 1; j < 16; ++j) {
      pr *= -cp[j0 + j - 1];
      oB[j * 16 + i] = pr;
    }
    for (int j = 0; j < i; ++j) oB[j * 16 + i] = 0.f;
  }
  float pc = 1.f;
  for (int i = 15; i >= 0; --i) {
    pc *= -cp[j0 + i];          // block 15: cp[255]==0 -> C==0
    oB[256 + i] = pc;
  }
}

// ---------------------------------------------------------------------------
// One sweep over 256 lines. Each wave owns line-groups {wv, wv+8} (16 lines
// each), interleaved for latency hiding. LS = LDS stride between lines,
// PS = stride along a line (X: 257/1, Y: 1/257).
// ---------------------------------------------------------------------------
__device__ __forceinline__ void run_sweep(float* __restrict__ U,
                                          const float* __restrict__ oF,
                                          const float* __restrict__ oB,
                                          const int LS, const int PS,
                                          const int tid) {
  const int lane = tid & 31;
  const int wv = tid >> 5;
  const int m = lane & 15;          // N index (column of C/D tile)
  const int h = lane >> 4;
  const int kh = 2 * h;             // K offset for this lane half (A & B ops)
  const int bidx15 = (((lane & 16) | 15) << 2);  // fwd carry: col 15 bcast
  const int bidx0 = ((lane & 16) << 2);          // bwd carry: col 0 bcast
  int line[2], st[2];
  line[0] = (wv * 16 + m) * LS;
  line[1] = ((wv + 8) * 16 + m) * LS;
  st[0] = (wv * 16 + 8 * h) * LS + m * PS;        // C/D tile: row = v + 8h
  st[1] = ((wv + 8) * 16 + 8 * h) * LS + m * PS;

  v8f prev[2];
#pragma unroll
  for (int g = 0; g < 2; ++g) prev[g] = v8f{0, 0, 0, 0, 0, 0, 0, 0};

  // ---- forward elimination: 16 blocks, carry chained left -> right ----
  for (int b = 0; b < 16; ++b) {
    const int j0 = b * 16;
    const float* op = oF + b * OPSZ;
    const float av = op[256 + m];   // A[n] per lane column
    v8f acc[2];
#pragma unroll
    for (int g = 0; g < 2; ++g)
#pragma unroll
      for (int v = 0; v < 8; ++v)
        acc[g][v] = bperm(bidx15, prev[g][v]) * av;  // C = carry (x) A
#pragma unroll
    for (int s4 = 0; s4 < 4; ++s4) {
      const int kab = 4 * s4 + kh;
      const int ko = kab * 16 + m;
      v2f bv;
      bv.x = op[ko];
      bv.y = op[ko + 16];
#pragma unroll
      for (int g = 0; g < 2; ++g) {
        v2f aa;
        aa.x = U[line[g] + (j0 + kab) * PS];
        aa.y = U[line[g] + (j0 + kab + 1) * PS];
        acc[g] = __builtin_amdgcn_wmma_f32_16x16x4_f32(
            false, aa, false, bv, (short)0, acc[g], false, false);
      }
    }
#pragma unroll
    for (int g = 0; g < 2; ++g) {
      prev[g] = acc[g];
#pragma unroll
      for (int v = 0; v < 8; ++v)
        U[st[g] + v * LS + j0 * PS] = acc[g][v];    // dp overwrites d
    }
  }

  // ---- back substitution: 16 blocks, carry chained right -> left ----
#pragma unroll
  for (int g = 0; g < 2; ++g) prev[g] = v8f{0, 0, 0, 0, 0, 0, 0, 0};
  for (int b = 15; b >= 0; --b) {
    const int j0 = b * 16;
    const float* op = oB + b * OPSZ;
    const float cv = op[256 + m];
    v8f acc[2];
#pragma unroll
    for (int g = 0; g < 2; ++g)
#pragma unroll
      for (int v = 0; v < 8; ++v)
        acc[g][v] = bperm(bidx0, prev[g][v]) * cv;
#pragma unroll
    for (int s4 = 0; s4 < 4; ++s4) {
      const int kab = 4 * s4 + kh;
      const int ko = kab * 16 + m;
      v2f bv;
      bv.x = op[ko];
      bv.y = op[ko + 16];
#pragma unroll
      for (int g = 0; g < 2; ++g) {
        v2f aa;
        aa.x = U[line[g] + (j0 + kab) * PS];
        aa.y = U[line[g] + (j0 + kab + 1) * PS];
        acc[g] = __builtin_amdgcn_wmma_f32_16x16x4_f32(
            false, aa, false, bv, (short)0, acc[g], false, false);
      }
    }
#pragma unroll
    for (int g = 0; g < 2; ++g) {
      prev[g] = acc[g];
#pragma unroll
      for (int v = 0; v < 8; ++v)
        U[st[g] + v * LS + j0 * PS] = acc[g][v];    // x overwrites dp
    }
  }
}

__device__ __forceinline__ void stage_run(float* U, float* opFl, float* opBl,
                                          const float* __restrict__ ws, int set,
                                          bool isX, int tid) {
  __syncthreads();  // previous sweep done before operator staging overwrites
  const float* gF = ws + (size_t)set * OPSET;
  const float* gB = ws + OPB_OFF + (size_t)set * OPSET;
  for (int i = tid; i < OPSET; i += 256) {
    opFl[i] = gF[i];
    opBl[i] = gB[i];
  }
  __syncthreads();
  if (isX)
    run_sweep(U, opFl, opBl, PITCH, 1, tid);
  else
    run_sweep(U, opFl, opBl, 1, PITCH, tid);
}

// ---------------------------------------------------------------------------
// Main kernel: 1 block per image, 256 threads (8 waves).
// ---------------------------------------------------------------------------
__global__ __launch_bounds__(256) void diffuse_main(
    const float* __restrict__ u_in, float* __restrict__ u_out,
    const float* __restrict__ ws) {
  extern __shared__ float lds[];
  float* U = lds;                        // 256 x PITCH image tile
  float* opFl = lds + SIZE * PITCH;      // staged fwd operators (4352 f)
  float* opBl = opFl + OPSET;            // staged bwd operators (4352 f)
  const int tid = threadIdx.x;
  const float* gsrc = u_in + (size_t)blockIdx.x * (SIZE * SIZE);

  // ---- image -> LDS, 257-float pitch ----
#if defined(__has_builtin) && __has_builtin(__builtin_amdgcn_tensor_load_to_lds)
  if (tid < 32) {  // one TDM op per workgroup (EXEC-independent DMA)
    const uint32_t ldsoff = (uint32_t)(size_t)(f32as3*)U;
    const uint64_t ga = (uint64_t)(size_t)gsrc;
    // D# group0: count=1, lds_addr, 57b global addr, type=2 ("image")
    v4u g0 = {1u, ldsoff, (uint32_t)ga, (uint32_t)(ga >> 32) | 0x80000000u};
    // D# group1: data_size=4B, pad_enable, pad_interval=256dw, pad_amount=1dw,
    // tensor_dim0=256, tensor_dim1=256, tile 256x256, dim0_stride=256.
    v8i g1 = {(int)0x01D20000, (int)0x01000000, (int)0x01000000,
              (int)0x01000000, 0x100, 256, 0, 1};
    v4i z4 = {0, 0, 0, 0};
#if defined(__clang_major__) && __clang_major__ >= 23
    v8i z8 = {0, 0, 0, 0, 0, 0, 0, 0};
    __builtin_amdgcn_tensor_load_to_lds(g0, g1, z4, z4, z8, 0);
#else
    __builtin_amdgcn_tensor_load_to_lds(g0, g1, z4, z4, 0);
#endif
    __builtin_amdgcn_s_wait_tensorcnt(0);
  }
  __syncthreads();
#else
  for (int it = 0; it < 64; ++it) {
    const int f = it * 256 + tid;                   // float4 index
    const float4 val = ((const float4*)gsrc)[f];
    const int row = f >> 6, c0 = (f & 63) * 4;
    float* dst = &U[row * PITCH + c0];
    dst[0] = val.x; dst[1] = val.y; dst[2] = val.z; dst[3] = val.w;
  }
  __syncthreads();
#endif

  // ---- 10 ADI steps: X(set k), Y(set k), X(set k+1) ----
  for (int k = 0; k < NSTEP; ++k) {
    stage_run(U, opFl, opBl, ws, k, true, tid);
    stage_run(U, opFl, opBl, ws, 11 + k, false, tid);
    stage_run(U, opFl, opBl, ws, k + 1, true, tid);
  }
  __syncthreads();

  // ---- LDS -> out ----
  float4* gout = (float4*)u_out + (size_t)blockIdx.x * (SIZE * SIZE / 4);
  for (int it = 0; it < 64; ++it) {
    const int f = it * 256 + tid;
    const int row = f >> 6, c0 = (f & 63) * 4;
    const float* src = &U[row * PITCH + c0];
    float4 val;
    val.x = src[0]; val.y = src[1]; val.z = src[2]; val.w = src[3];
    gout[f] = val;
  }
}

// ---------------------------------------------------------------------------
extern "C" void kernel_launch(void* const* d_in, const int* in_sizes, int n_in,
                              void* d_out, int out_size, void* d_ws,
                              size_t ws_size, hipStream_t stream) {
  const float* u    = (const float*)d_in[0];
  const float* abx  = (const float*)d_in[1];   // alpha_base_x
  const float* bby  = (const float*)d_in[4];   // beta_base_y
  const float* atcx = (const float*)d_in[5];   // alpha_time_coeff_x
  const float* btcy = (const float*)d_in[8];   // beta_time_coeff_y
  const float* atqx = (const float*)d_in[9];   // alpha_time_quad_x
  const float* btqy = (const float*)d_in[12];  // beta_time_quad_y
  float* ws = (float*)d_ws;                    // needs ~796KB
  const int B = in_sizes[0] / (SIZE * SIZE);

  setup_scal<<<1, 32, 0, stream>>>(abx, atcx, atqx, bby, btcy, btqy, ws);
  setup_ops<<<2, 256, 0, stream>>>(ws);
  (void)hipFuncSetAttribute((const void*)diffuse_main,
                            hipFuncAttributeMaxDynamicSharedMemorySize,
                            SMEM_BYTES);
  diffuse_main<<<B, 256, SMEM_BYTES, stream>>>(u, (float*)d_out, ws);
}